// PermutationMatrixCalculator_56779467653261
// MI455X (gfx1250) — compile-verified
//
#include <hip/hip_runtime.h>
#include <stdint.h>

#define NROWS 16384
#define K     128
#define WAVES 4
#define LANES 32

// Get the raw LDS byte offset of a __shared__ object (addrspacecast generic->AS3,
// then ptrtoint; hardware adds LDS_BASE in the async-store path).
__device__ __forceinline__ uint32_t lds_offset(const void* p) {
  return (uint32_t)(uintptr_t)(const __attribute__((address_space(3))) void*)p;
}

// rank += (v > xe) || (v == xe && m < je)   [stable descending argsort rank]
#define CMP(val, midx, xe, je, acc) \
  (acc) += (int)((val) > (xe)) | (int)(((val) == (xe)) & ((midx) < (je)));

__global__ __launch_bounds__(WAVES * LANES)
void PermutationMatrixCalculator_56779467653261_kernel(const float* __restrict__ x,
                                                       float* __restrict__ out) {
  // 5 one-hot payload patterns of 4 floats each: pat[0]=zeros, pat[1+e] has 1.0 at e.
  __shared__ __align__(16) float pat[5][4];
  __shared__ __align__(16) float rowbuf[WAVES][K];
  __shared__ __align__(16) int   inv[WAVES][K];

  const int tid  = threadIdx.x;
  const int lane = tid & (LANES - 1);
  const int w    = tid >> 5;
  const int n    = blockIdx.x * WAVES + w;

  if (tid < 20) {
    const int r = tid >> 2, c = tid & 3;
    pat[r][c] = (r == c + 1) ? 1.0f : 0.0f;
  }

  // ---- load row: 4 consecutive elements per lane (coalesced b128) ----
  const float4 xv = reinterpret_cast<const float4*>(x + (size_t)n * K)[lane];
  reinterpret_cast<float4*>(&rowbuf[w][0])[lane] = xv;
  __syncthreads();

  // ---- ranks via all-to-all comparison (broadcast LDS reads) ----
  const int j0 = lane * 4;
  const int j1 = j0 + 1, j2 = j0 + 2, j3 = j0 + 3;
  int r0 = 0, r1 = 0, r2 = 0, r3 = 0;
  const float4* rb4 = reinterpret_cast<const float4*>(&rowbuf[w][0]);
#pragma unroll 4
  for (int m4 = 0; m4 < K / 4; ++m4) {
    const float4 v = rb4[m4];
    const int m = m4 * 4;
    CMP(v.x, m + 0, xv.x, j0, r0) CMP(v.y, m + 1, xv.x, j0, r0)
    CMP(v.z, m + 2, xv.x, j0, r0) CMP(v.w, m + 3, xv.x, j0, r0)
    CMP(v.x, m + 0, xv.y, j1, r1) CMP(v.y, m + 1, xv.y, j1, r1)
    CMP(v.z, m + 2, xv.y, j1, r1) CMP(v.w, m + 3, xv.y, j1, r1)
    CMP(v.x, m + 0, xv.z, j2, r2) CMP(v.y, m + 1, xv.z, j2, r2)
    CMP(v.z, m + 2, xv.z, j2, r2) CMP(v.w, m + 3, xv.z, j2, r2)
    CMP(v.x, m + 0, xv.w, j3, r3) CMP(v.y, m + 1, xv.w, j3, r3)
    CMP(v.z, m + 2, xv.w, j3, r3) CMP(v.w, m + 3, xv.w, j3, r3)
  }

  // inverse permutation: inv[rank[j]] = j  (ranks are a permutation -> no collisions)
  inv[w][r0] = j0; inv[w][r1] = j1; inv[w][r2] = j2; inv[w][r3] = j3;
  __syncthreads();

  // ---- stream the K x K one-hot matrix out via async LDS->global DMA ----
  // Lane L owns columns [4L, 4L+4). For output row i with col = inv[i], lane
  // col>>2 sources pattern 1+(col&3); all other lanes source the zero pattern.
  // Non-temporal: 1 GiB pure write stream, no reuse -> don't retain in caches.
  const uint32_t pat_base = lds_offset(&pat[0][0]);
  float* orow = out + ((size_t)n << 14);  // n * K * K
  const int4* inv4 = reinterpret_cast<const int4*>(&inv[w][0]);

#define EMIT_ROW(i, col)                                                        \
  {                                                                             \
    const uint32_t lp = pat_base +                                              \
        ((((col) >> 2) == lane) ? (uint32_t)((1 + ((col) & 3)) << 4) : 0u);     \
    const float* g = orow + ((size_t)(i) << 7) + (j0);                          \
    asm volatile("global_store_async_from_lds_b128 %0, %1, off th:TH_STORE_NT"  \
                 :: "v"(g), "v"(lp) : "memory");                                \
  }

  for (int i4 = 0; i4 < K / 4; ++i4) {
    const int4 c = inv4[i4];   // broadcast read of 4 target columns
    const int i = i4 * 4;
    EMIT_ROW(i + 0, c.x)
    EMIT_ROW(i + 1, c.y)
    EMIT_ROW(i + 2, c.z)
    EMIT_ROW(i + 3, c.w)
  }

  // Drain async stores (S_ENDPGM also implies wait-idle; explicit for clarity).
  asm volatile("s_wait_asynccnt 0" ::: "memory");
}

extern "C" void kernel_launch(void* const* d_in, const int* in_sizes, int n_in,
                              void* d_out, int out_size, void* d_ws, size_t ws_size,
                              hipStream_t stream) {
  (void)in_sizes; (void)n_in; (void)d_ws; (void)ws_size; (void)out_size;
  const float* x = (const float*)d_in[0];
  float* out = (float*)d_out;
  dim3 grid(NROWS / WAVES), block(WAVES * LANES);
  hipLaunchKernelGGL(PermutationMatrixCalculator_56779467653261_kernel,
                     grid, block, 0, stream, x, out);
}